// Quanv2d_19078244728891
// MI455X (gfx1250) — compile-verified
//
#include <hip/hip_runtime.h>

typedef __attribute__((ext_vector_type(2))) float v2f;
typedef __attribute__((ext_vector_type(8))) float v8f;

#define PSTRIDE 17   // 16 probs + 1 pad -> conflict-free LDS banks

__global__ __launch_bounds__(256) void quanv_kernel(const float* __restrict__ x,
                                                    const float* __restrict__ w,
                                                    float* __restrict__ out)
{
    __shared__ float probsLDS[256 * PSTRIDE];  // per-patch |amp|^2 staging
    __shared__ float outLDS[256 * 4];          // per-patch expvals staging

    const int tid  = threadIdx.x;
    const int lane = tid & 31;
    const int wave = tid >> 5;
    const int p    = blockIdx.x * 256 + tid;   // global patch id

    // ---- load 2x2 patch: two coalesced float2 loads ----
    const int b  = p >> 12;          // image  (4096 patches / image)
    const int pi = (p >> 6) & 63;    // patch row
    const int pj = p & 63;           // patch col
    const float2* rp = reinterpret_cast<const float2*>(
        x + ((size_t)b * 128 + 2 * pi) * 128) + pj;
    float2 a01 = rp[0];
    float2 a23 = rp[64];             // next image row (+128 floats)
    float ang[4] = { a01.x, a01.y, a23.x, a23.y };

    // ---- initial product state |psi> = RY(a_q) H |0> per qubit (real) ----
    const float inv_sqrt2 = 0.70710678118654752440f;
    float v[4][2];
#pragma unroll
    for (int q = 0; q < 4; ++q) {
        float s, c;
        __sincosf(0.5f * ang[q], &s, &c);
        v[q][0] = (c - s) * inv_sqrt2;
        v[q][1] = (c + s) * inv_sqrt2;
    }
    float sr[16], si[16];
#pragma unroll
    for (int idx = 0; idx < 16; ++idx) {
        sr[idx] = v[0][(idx >> 3) & 1] * v[1][(idx >> 2) & 1] *
                  v[2][(idx >> 1) & 1] * v[3][idx & 1];
        si[idx] = 0.0f;
    }

    // ---- 2 layers: fused CRZ ring diagonal + per-qubit RY ----
#pragma unroll
    for (int layer = 0; layer < 2; ++layer) {
        const float* wl = w + layer * 8;
        // CRZ(phi_i): control qubit i, target qubit (i+1)%4.
        // Apply as 4 per-qubit complex factors (4 sincos, not 16).
#pragma unroll
        for (int i = 0; i < 4; ++i) {
            float ph_s, ph_c;
            __sincosf(0.5f * wl[i], &ph_s, &ph_c);
            const int cb = 3 - i;               // control bit position
            const int tb = 3 - ((i + 1) & 3);   // target bit position
#pragma unroll
            for (int idx = 0; idx < 16; ++idx) {
                if ((idx >> cb) & 1) {
                    float ss = ((idx >> tb) & 1) ? ph_s : -ph_s;
                    float nr = sr[idx] * ph_c - si[idx] * ss;
                    float ni = sr[idx] * ss + si[idx] * ph_c;
                    sr[idx] = nr; si[idx] = ni;
                }
            }
        }
        // RY(theta_q) on each qubit: (x0,x1) -> (c x0 - s x1, s x0 + c x1)
#pragma unroll
        for (int q = 0; q < 4; ++q) {
            float st, ct;
            __sincosf(0.5f * wl[4 + q], &st, &ct);
            const int bit = 1 << (3 - q);
#pragma unroll
            for (int idx = 0; idx < 16; ++idx) {
                if (!(idx & bit)) {
                    const int j = idx | bit;
                    float r0 = sr[idx], i0 = si[idx];
                    float r1 = sr[j],   i1 = si[j];
                    sr[idx] = ct * r0 - st * r1;  si[idx] = ct * i0 - st * i1;
                    sr[j]   = st * r0 + ct * r1;  si[j]   = st * i0 + ct * i1;
                }
            }
        }
    }

    // ---- probabilities -> LDS (stride 17: conflict-free) ----
    float* myProbs = &probsLDS[tid * PSTRIDE];
#pragma unroll
    for (int idx = 0; idx < 16; ++idx)
        myProbs[idx] = sr[idx] * sr[idx] + si[idx] * si[idx];

    __syncthreads();

    // ---- expvals = probs[16x16] @ ZMAT[16x4 (padded to 16)] ----
    // per wave: two 16-patch groups, each = 4 chained V_WMMA_F32_16X16X4_F32
    const int hb = lane >> 4;    // half-wave selector
    const int ln = lane & 15;
#pragma unroll
    for (int g = 0; g < 2; ++g) {
        const int mbase = wave * 32 + g * 16;   // first patch of group (in block)
        v8f acc = {};
#pragma unroll
        for (int k = 0; k < 4; ++k) {
            const int kk = 4 * k + 2 * hb;      // K index pair per A/B VGPR layout
            // A 16x4 f32: lanes0-15 K={4k,4k+1}, lanes16-31 K={4k+2,4k+3}, M=ln
            v2f a;
            a.x = probsLDS[(mbase + ln) * PSTRIDE + kk];
            a.y = probsLDS[(mbase + ln) * PSTRIDE + kk + 1];
            // B 4x16: row kk / kk+1 striped across lanes; Z[s][q]=1-2*bit(3-q)(s)
            v2f bm;
            bm.x = (ln < 4) ? (1.0f - 2.0f * (float)((kk       >> (3 - ln)) & 1)) : 0.0f;
            bm.y = (ln < 4) ? (1.0f - 2.0f * (float)(((kk + 1) >> (3 - ln)) & 1)) : 0.0f;
            acc = __builtin_amdgcn_wmma_f32_16x16x4_f32(false, a, false, bm,
                                                        (short)0, acc, false, false);
        }
        // D layout: VGPR r = (M=r, N=ln | hb=0), (M=r+8, N=ln | hb=1)
        if (ln < 4) {
#pragma unroll
            for (int r = 0; r < 8; ++r) {
                const int m = mbase + r + 8 * hb;
                outLDS[m * 4 + ln] = acc[r];
            }
        }
    }

    __syncthreads();

    // ---- coalesced contiguous store: 256 patches x float4 per block ----
    float4* o4 = reinterpret_cast<float4*>(out) + (size_t)blockIdx.x * 256 + tid;
    *o4 = reinterpret_cast<const float4*>(outLDS)[tid];
}

extern "C" void kernel_launch(void* const* d_in, const int* in_sizes, int n_in,
                              void* d_out, int out_size, void* d_ws, size_t ws_size,
                              hipStream_t stream) {
    const float* x = (const float*)d_in[0];   // [B,1,128,128] f32
    const float* w = (const float*)d_in[1];   // [2,8] f32
    float* out = (float*)d_out;               // [B, 4096*4] f32

    const int B = in_sizes[0] / (128 * 128);
    const int patches = B * 64 * 64;
    const int blocks = patches / 256;         // 256 patches (threads) per block
    quanv_kernel<<<blocks, 256, 0, stream>>>(x, w, out);
}